// GenerateHeatmap_88940182765706
// MI455X (gfx1250) — compile-verified
//
#include <hip/hip_runtime.h>
#include <hip/hip_bf16.h>

// GenerateHeatmap on MI455X (gfx1250, wave32).
// out[b][kp][y][x] = exp(-((x-cx)^2 + (y-cy)^2) / (2*sigma^2))
// Separable Gaussian -> each 16x16 tile is a rank-1 outer product ey (x) ex,
// computed with one V_WMMA_F32_16X16X4_F32 (A col K=0 = ey, B row K=0 = ex).
// Bandwidth bound (1.14 GB out @ 23.3 TB/s ~= 49 us floor); stores are
// non-temporal since output >> 192 MB L2.

#define IMG_H 256
#define IMG_W 128
#define N_KP  17
// 1/(2*sigma^2) * log2(e), sigma = 6.0
#define K2F   (0.0138888889f * 1.4426950408889634f)

typedef __attribute__((ext_vector_type(2))) float v2f;
typedef __attribute__((ext_vector_type(8))) float v8f;

__global__ __launch_bounds__(256) void
heatmap_wmma_kernel(const float* __restrict__ x, float* __restrict__ out)
{
    // 8 waves per block, one 16x16 tile per wave.
    const int wave  = (blockIdx.x << 3) + (threadIdx.x >> 5);
    const int lane  = threadIdx.x & 31;
    const int half  = lane >> 4;      // 0: lanes 0-15, 1: lanes 16-31
    const int l     = lane & 15;

    // Tile decomposition: 128 tiles per (b,kp) image: 16 (y) x 8 (x).
    const int image = wave >> 7;
    const int t     = wave & 127;
    const int y0    = (t >> 3) << 4;  // tile row * 16
    const int x0    = (t & 7) << 4;   // tile col * 16

    const int b  = image / N_KP;
    const int kp = image - b * N_KP;

    // x layout: (B, 2*N_KP): row = [cx(0..16), cy(0..16)]
    const float cx = x[b * (2 * N_KP) + kp];
    const float cy = x[b * (2 * N_KP) + N_KP + kp];

    // Separable 1-D Gaussians (one v_exp_f32 each).
    const float fy = (float)(y0 + l) - cy;
    const float fx = (float)(x0 + l) - cx;
    const float ey = __builtin_amdgcn_exp2f(-(fy * fy) * K2F);
    const float ex = __builtin_amdgcn_exp2f(-(fx * fx) * K2F);

    // A: 16x4 f32. VGPR0 = {K=0 in lanes 0-15, K=2 in lanes 16-31}, VGPR1 = {K=1,K=3}.
    // Only K=0 (lanes 0-15 of a.x) is nonzero -> rank-1 update.
    v2f a; a.x = half ? 0.0f : ey; a.y = 0.0f;
    v2f bm; bm.x = half ? 0.0f : ex; bm.y = 0.0f;
    v8f c = {};

    // D = A x B + 0 : full 16x16 tile of ey[m]*ex[n].
    v8f d = __builtin_amdgcn_wmma_f32_16x16x4_f32(
        /*neg_a=*/false, a, /*neg_b=*/false, bm,
        /*c_mod=*/(short)0, c, /*reuse_a=*/false, /*reuse_b=*/false);

    // D layout: VGPR i -> row (y0+i) in lanes 0-15, row (y0+i+8) in lanes 16-31,
    // column = x0 + (lane & 15). Two contiguous 64B segments per store.
    float* base = out + (size_t)image * (IMG_H * IMG_W)
                      + (size_t)(y0 + half * 8) * IMG_W + (x0 + l);
#pragma unroll
    for (int i = 0; i < 8; ++i) {
        __builtin_nontemporal_store(d[i], base + i * IMG_W);
    }
}

extern "C" void kernel_launch(void* const* d_in, const int* in_sizes, int n_in,
                              void* d_out, int out_size, void* d_ws, size_t ws_size,
                              hipStream_t stream)
{
    const float* x = (const float*)d_in[0];
    float* out = (float*)d_out;

    // out_size = B * N_KP * H * W; each image has 128 tiles; 8 waves per block.
    const int nimg   = out_size / (IMG_H * IMG_W);
    const int tiles  = nimg * ((IMG_H / 16) * (IMG_W / 16)); // multiple of 128
    const int blocks = tiles >> 3;                            // exact (tiles % 8 == 0)

    heatmap_wmma_kernel<<<blocks, 256, 0, stream>>>(x, out);
}